// SGConv_24850680775444
// MI455X (gfx1250) — compile-verified
//
#include <hip/hip_runtime.h>

typedef __attribute__((ext_vector_type(2))) float v2f;
typedef __attribute__((ext_vector_type(8))) float v8f;

#define DIN 128
#define COLS 512           // SIZE * D_OUT
#define KTOT 256           // 2 * DIN  (mean-part + x-part)
#define AS_STRIDE 260      // 256 + 4 pad -> conflict-free ds_load_b64 across 16 lanes

// Scheduling-pipeline pragmas (IGroupLP). Masks: 0x8 MFMA/WMMA, 0x20 VMEM read,
// 0x100 DS read.
#if defined(__has_builtin)
#if __has_builtin(__builtin_amdgcn_sched_group_barrier)
#define SGB(mask, size, id) __builtin_amdgcn_sched_group_barrier((mask), (size), (id))
#else
#define SGB(mask, size, id)
#endif
#else
#define SGB(mask, size, id)
#endif

// ---------------------------------------------------------------------------
// Kernel 1: zero the aggregation scratch (agg[N][128] + cnt[N]).
// ---------------------------------------------------------------------------
__global__ __launch_bounds__(256) void zero_ws(float* __restrict__ agg, int ast,
                                               float* __restrict__ cnt, int cst, int n) {
  int idx = blockIdx.x * 256 + threadIdx.x;
  int node = idx >> 7;
  int c = idx & 127;
  if (node < n) {
    agg[(size_t)node * ast + c] = 0.0f;
    if (c == 0) cnt[(size_t)node * cst] = 0.0f;
  }
}

// ---------------------------------------------------------------------------
// Kernel 2: prepack [W_l ; W_r] (virtual 256 x 512) into WMMA-friendly
// Bp[pair][col][2]: lane's (K, K+1) B-fragment pair becomes one b64 load,
// 16 consecutive lanes -> 128B contiguous. 512KB, one-time, L2-resident.
// ---------------------------------------------------------------------------
__global__ __launch_bounds__(256) void pack_w(const float* __restrict__ Wl,
                                              const float* __restrict__ Wr,
                                              float* __restrict__ Bp) {
  int idx = blockIdx.x * 256 + threadIdx.x;   // pair*512 + col, pairs=128
  if (idx >= (KTOT / 2) * COLS) return;
  int p = idx >> 9;        // K pair index: rows 2p, 2p+1
  int c = idx & 511;       // output column
  int s = c >> 7;
  int o = c & 127;
  int r0 = 2 * p;          // r0 even -> r0, r0+1 in same (W_l | W_r) half
  const float* W = (r0 < DIN) ? Wl : Wr;
  int rr = r0 & 127;
  v2f v;
  v.x = W[(size_t)s * DIN * 128 + (size_t)rr * 128 + o];
  v.y = W[(size_t)s * DIN * 128 + (size_t)(rr + 1) * 128 + o];
  *(v2f*)&Bp[(size_t)idx * 2] = v;
}

// ---------------------------------------------------------------------------
// Kernel 3: edge scatter-add. One wave32 per edge: 512B row gathered as one
// global_load_b128 per lane, scattered with hardware global_atomic_add_f32.
// ---------------------------------------------------------------------------
__global__ __launch_bounds__(256) void edge_agg(const float* __restrict__ x,
                                                const int* __restrict__ ei,
                                                float* __restrict__ agg, int ast,
                                                float* __restrict__ cnt, int cst, int E) {
  int e = blockIdx.x * 8 + (threadIdx.x >> 5);     // wave-uniform
  if (e >= E) return;
  int lane = threadIdx.x & 31;
  int src = ei[e];          // edge_index row 0
  int dst = ei[E + e];      // edge_index row 1
  const float4 v = *(const float4*)(x + (size_t)src * DIN + lane * 4);
  float* ap = agg + (size_t)dst * ast + lane * 4;
  __hip_atomic_fetch_add(ap + 0, v.x, __ATOMIC_RELAXED, __HIP_MEMORY_SCOPE_AGENT);
  __hip_atomic_fetch_add(ap + 1, v.y, __ATOMIC_RELAXED, __HIP_MEMORY_SCOPE_AGENT);
  __hip_atomic_fetch_add(ap + 2, v.z, __ATOMIC_RELAXED, __HIP_MEMORY_SCOPE_AGENT);
  __hip_atomic_fetch_add(ap + 3, v.w, __ATOMIC_RELAXED, __HIP_MEMORY_SCOPE_AGENT);
  if (lane == 0)
    __hip_atomic_fetch_add(cnt + (size_t)dst * cst, 1.0f, __ATOMIC_RELAXED,
                           __HIP_MEMORY_SCOPE_AGENT);
}

// ---------------------------------------------------------------------------
// Kernel 4: fused [mean || x] x [W_l ; W_r] + b, leaky_relu, via f32 WMMA.
// One block = 16 output rows x 512 cols. 8 waves x 4 col-tiles each.
// launch_bounds(256, 4): cap occupancy at 4 waves/EU so the scheduler keeps
// prefetched B fragments live instead of sinking loads to their consumers.
// sched_group_barrier pins the per-k-step order: 4 loads -> 1 DS -> 4 WMMA,
// where the WMMAs consume the PREVIOUS step's loads (cur/nxt double buffer).
// ---------------------------------------------------------------------------
template <bool PACKED>
__global__ __launch_bounds__(256, 4) void sage_gemm(
    const float* __restrict__ agg, int ast, const float* __restrict__ cnt, int cst,
    const float* __restrict__ x, const float* __restrict__ Wl,
    const float* __restrict__ Wr, const float* __restrict__ Bp,
    const float* __restrict__ bias, float* __restrict__ out, int n) {
  __shared__ float As[16 * AS_STRIDE];   // 16 rows x (mean:0..127 | x:128..255)

  const int g0 = blockIdx.x * 16;

  // --- stage A tile into LDS, fusing the mean divide ---
  for (int idx = threadIdx.x; idx < 16 * 256; idx += 256) {
    int m = idx >> 8;
    int c = idx & 255;
    int g = g0 + m;
    float val = 0.0f;
    if (g < n) {
      if (c < DIN) {
        float cv = cnt[(size_t)g * cst];
        val = agg[(size_t)g * ast + c] / fmaxf(cv, 1.0f);
      } else {
        val = x[(size_t)g * DIN + (c - DIN)];
      }
    }
    As[m * AS_STRIDE + c] = val;
  }
  __syncthreads();   // also fences the fallback path before we overwrite d_out

  const int lane = threadIdx.x & 31;
  const int wave = threadIdx.x >> 5;
  const int l16  = lane & 15;
  const int half = lane >> 4;
  const int kb   = half * 2;   // K offset of this lane's A/B fragment pair

  int col[4];
  v8f acc[4];
  const float* pB[4];   // packed path: per-ct base into Bp
  const float* Bl[4];   // unpacked path
  const float* Br[4];
#pragma unroll
  for (int ct = 0; ct < 4; ++ct) {
    int c = (wave * 4 + ct) * 16 + l16;
    col[ct] = c;
    int s = c >> 7;
    int o = c & 127;
    if (PACKED) {
      // Bp[pair][col][2]; this lane's pair at k0 is k0/2 + half.
      pB[ct] = Bp + (size_t)half * (COLS * 2) + (size_t)c * 2;
    } else {
      Bl[ct] = Wl + ((size_t)s * DIN * 128) + o;
      Br[ct] = Wr + ((size_t)s * DIN * 128) + o;
    }
    float bb = bias[c];      // b[s][o] == bias[s*128+o] == bias[c]
    acc[ct] = (v8f){bb, bb, bb, bb, bb, bb, bb, bb};
  }

  if (PACKED) {
    // Software-pipelined K loop over all 256 K-values.
    // Per k0 step: one float2 of B per ct at pB[ct] + k0*512 floats.
    v2f cur[4], nxt[4];
#pragma unroll
    for (int ct = 0; ct < 4; ++ct)
      cur[ct] = *(const v2f*)(pB[ct]);

#pragma unroll 4
    for (int k0 = 0; k0 < KTOT - 4; k0 += 4) {
      // prefetch next k-step's B fragments (independent dest values)
#pragma unroll
      for (int ct = 0; ct < 4; ++ct)
        nxt[ct] = *(const v2f*)(pB[ct] + (size_t)(k0 + 4) * (COLS / 2) * 2);
      v2f a = *(const v2f*)&As[l16 * AS_STRIDE + k0 + kb];   // ds_load_b64
#pragma unroll
      for (int ct = 0; ct < 4; ++ct) {
        acc[ct] = __builtin_amdgcn_wmma_f32_16x16x4_f32(
            false, a, false, cur[ct], (short)0, acc[ct], false, false);
        cur[ct] = nxt[ct];
      }
      // Pipeline shape for this k-step: issue the 4 prefetch loads first,
      // then the A-fragment DS read, then the 4 WMMAs (which consume the
      // previous step's loads) -> keeps >=4 b64 loads in flight.
      SGB(0x020, 4, 0);   // 4x VMEM read
      SGB(0x100, 1, 0);   // 1x DS read
      SGB(0x008, 4, 0);   // 4x WMMA
    }
    {  // epilogue: last k-step
      v2f a = *(const v2f*)&As[l16 * AS_STRIDE + (KTOT - 4) + kb];
#pragma unroll
      for (int ct = 0; ct < 4; ++ct)
        acc[ct] = __builtin_amdgcn_wmma_f32_16x16x4_f32(
            false, a, false, cur[ct], (short)0, acc[ct], false, false);
    }
  } else {
#pragma unroll 4
    for (int k0 = 0; k0 < DIN; k0 += 4) {
      int kk = k0 + kb;
      v2f a = *(const v2f*)&As[l16 * AS_STRIDE + kk];
#pragma unroll
      for (int ct = 0; ct < 4; ++ct) {
        v2f bf;
        bf.x = Bl[ct][(size_t)kk * 128];
        bf.y = Bl[ct][(size_t)(kk + 1) * 128];
        acc[ct] = __builtin_amdgcn_wmma_f32_16x16x4_f32(
            false, a, false, bf, (short)0, acc[ct], false, false);
      }
    }
#pragma unroll 4
    for (int k0 = 0; k0 < DIN; k0 += 4) {
      int kk = k0 + kb;
      v2f a = *(const v2f*)&As[l16 * AS_STRIDE + DIN + kk];
#pragma unroll
      for (int ct = 0; ct < 4; ++ct) {
        v2f bf;
        bf.x = Br[ct][(size_t)kk * 128];
        bf.y = Br[ct][(size_t)(kk + 1) * 128];
        acc[ct] = __builtin_amdgcn_wmma_f32_16x16x4_f32(
            false, a, false, bf, (short)0, acc[ct], false, false);
      }
    }
  }

  // --- leaky_relu + store (lane owns one column, 8 rows) ---
  const int row0 = g0 + half * 8;
#pragma unroll
  for (int ct = 0; ct < 4; ++ct) {
    float* op = out + (size_t)row0 * COLS + col[ct];
#pragma unroll
    for (int i = 0; i < 8; ++i) {
      if (row0 + i < n) {
        float v = acc[ct][i];
        v = v > 0.0f ? v : 0.01f * v;
        op[(size_t)i * COLS] = v;
      }
    }
  }
}

// ---------------------------------------------------------------------------
extern "C" void kernel_launch(void* const* d_in, const int* in_sizes, int n_in,
                              void* d_out, int out_size, void* d_ws, size_t ws_size,
                              hipStream_t stream) {
  const float* x    = (const float*)d_in[0];
  const int*   ei   = (const int*)d_in[1];
  const float* Wl   = (const float*)d_in[2];
  const float* Wr   = (const float*)d_in[3];
  const float* bias = (const float*)d_in[4];
  float* out = (float*)d_out;

  const int N = in_sizes[0] / DIN;     // 50000
  const int E = in_sizes[1] / 2;       // 800000

  const size_t aggFloats = (size_t)N * DIN + (size_t)N;          // agg + cnt
  const size_t aggBytes  = aggFloats * sizeof(float);
  const size_t bpBytes   = (size_t)KTOT * COLS * sizeof(float);  // 512 KB

  // Tiered scratch selection (deterministic, host-side):
  float* agg;
  float* cnt;
  int ast, cst;
  float* Bp = nullptr;
  if (ws_size >= aggBytes + bpBytes) {
    agg = (float*)d_ws;                     ast = DIN;
    cnt = (float*)d_ws + (size_t)N * DIN;   cst = 1;
    Bp  = (float*)d_ws + aggFloats;         // 8B-aligned (aggFloats*4 % 8 == 0)
  } else if (ws_size >= aggBytes) {
    agg = (float*)d_ws;                     ast = DIN;
    cnt = (float*)d_ws + (size_t)N * DIN;   cst = 1;
  } else if (ws_size >= bpBytes) {
    Bp  = (float*)d_ws;
    agg = out;                              ast = COLS;
    cnt = out + DIN;                        cst = COLS;
  } else {
    agg = out;                              ast = COLS;
    cnt = out + DIN;                        cst = COLS;
  }

  dim3 blk(256);
  int zero_blocks = (int)(((size_t)N * DIN + 255) / 256);
  zero_ws<<<zero_blocks, blk, 0, stream>>>(agg, ast, cnt, cst, N);

  if (Bp) {
    int pack_blocks = ((KTOT / 2) * COLS + 255) / 256;   // 256 blocks
    pack_w<<<pack_blocks, blk, 0, stream>>>(Wl, Wr, Bp);
  }

  int edge_blocks = (E + 7) / 8;
  edge_agg<<<edge_blocks, blk, 0, stream>>>(x, ei, agg, ast, cnt, cst, E);

  int gemm_blocks = (N + 15) / 16;
  if (Bp) {
    sage_gemm<true><<<gemm_blocks, blk, 0, stream>>>(agg, ast, cnt, cst, x, Wl, Wr,
                                                     Bp, bias, out, N);
  } else {
    sage_gemm<false><<<gemm_blocks, blk, 0, stream>>>(agg, ast, cnt, cst, x, Wl, Wr,
                                                      Bp, bias, out, N);
  }
}